// IDWTLayer_12077448037070
// MI455X (gfx1250) — compile-verified
//
#include <hip/hip_runtime.h>

typedef float v2f __attribute__((ext_vector_type(2)));
typedef float v8f __attribute__((ext_vector_type(8)));

#define S_IN   8192
#define S_OUT  16384
#define C_HALF 64
#define C_IN   128

// out[w0+r, ch] tile = A_lo(16x12) * Xa(12x16) + A_hi(16x12) * Xd(12x16)
// A_lo[r][k] = lo[2*(k - (r>>1)) + ((r&1)^1)] for (k - (r>>1)) in [0,4), else 0.
// K chunked by 4 -> 3 chunks per filter -> 6 v_wmma_f32_16x16x4_f32 per 16x16 tile.
__global__ __launch_bounds__(256) void idwt_db4_wmma_kernel(
    const float* __restrict__ x,       // (16, 8192, 128)
    const float* __restrict__ rec_lo,  // (8,)
    const float* __restrict__ rec_hi,  // (8,)
    float* __restrict__ out)           // (16, 16384, 64)
{
    const int lane = threadIdx.x & 31;
    const int wid  = blockIdx.x * (blockDim.x >> 5) + (threadIdx.x >> 5);
    const int b    = wid >> 10;        // 16 batches
    const int p    = wid & 1023;       // 1024 w-tiles of 16 outputs
    const int w0   = p << 4;
    const int j0   = (p << 3) - 1;     // input row for K=0 (m-1 of first even output)

    const int r    = lane & 15;        // A-row / B,D column
    const int hi16 = lane >> 4;
    const int kk   = hi16 << 1;        // K sub-offset within a chunk: {0, 2}
    const int sel  = (r & 1) ^ 1;      // even w -> odd taps, odd w -> even taps
    const int base = r >> 1;

    // Filters are uniform across the wave -> scalar loads.
    float lo[8], hf[8];
#pragma unroll
    for (int i = 0; i < 8; ++i) { lo[i] = rec_lo[i]; hf[i] = rec_hi[i]; }

    // Constant band-matrix A fragments (built once, reused for all 4 channel tiles).
    v2f Alo[3], Ahi[3];
#pragma unroll
    for (int c = 0; c < 3; ++c) {
#pragma unroll
        for (int s = 0; s < 2; ++s) {
            const int k = 4 * c + kk + s;
            const int d = k - base;                 // tap index within band
            const bool in = (d >= 0) && (d < 4);    // implies k <= 10 (K=11 pad column -> 0)
            float vlo = in ? ((d == 0) ? lo[sel] : (d == 1) ? lo[2 + sel]
                            : (d == 2) ? lo[4 + sel] : lo[6 + sel]) : 0.0f;
            float vhi = in ? ((d == 0) ? hf[sel] : (d == 1) ? hf[2 + sel]
                            : (d == 2) ? hf[4 + sel] : hf[6 + sel]) : 0.0f;
            Alo[c][s] = vlo;
            Ahi[c][s] = vhi;
        }
    }

    const float* xa = x + (size_t)b * S_IN * C_IN;  // approx: channels [0,64)
    float* ob = out + (size_t)b * S_OUT * C_HALF;

    // Per-lane store base: row w0 + 8*hi16, channel r. All store offsets immediate.
    float* po = ob + (size_t)(w0 + (hi16 << 3)) * C_HALF + r;

    if (p != 0 && p != 1023) {
        // Interior fast path: rows j0+kk .. j0+kk+9 all within [0, 8192).
        // Single base pointer; every load offset is a compile-time immediate:
        //   chunk c -> +c*512 floats, s=1 -> +128, detail -> +64, ct -> +16.
        const float* pb = xa + (size_t)(j0 + kk) * C_IN + r;
#pragma unroll
        for (int ct = 0; ct < 4; ++ct) {
            const int co = ct << 4;                 // channel-tile offset (floats)
            v8f acc = {};
#pragma unroll
            for (int c = 0; c < 3; ++c) {
                const int o = co + c * 512;
                v2f Ba, Bd;
                Ba[0] = pb[o];
                Ba[1] = pb[o + 128];
                Bd[0] = pb[o + 64];
                Bd[1] = pb[o + 192];
                acc = __builtin_amdgcn_wmma_f32_16x16x4_f32(
                    false, Alo[c], false, Ba, (short)0, acc, false, false);
                acc = __builtin_amdgcn_wmma_f32_16x16x4_f32(
                    false, Ahi[c], false, Bd, (short)0, acc, false, false);
            }
#pragma unroll
            for (int v = 0; v < 8; ++v) po[co + v * 64] = acc[v];
        }
    } else {
        // Boundary tiles (p==0 or p==1023): clamp row to a valid address,
        // load unconditionally, zero via select (no exec-mask branches).
#pragma unroll
        for (int ct = 0; ct < 4; ++ct) {
            const int co = ct << 4;
            const int ch = co + r;
            v8f acc = {};
#pragma unroll
            for (int c = 0; c < 3; ++c) {
                v2f Ba, Bd;
#pragma unroll
                for (int s = 0; s < 2; ++s) {
                    const int jj = j0 + 4 * c + kk + s;
                    const bool ok = (jj >= 0) && (jj < S_IN);   // SAME zero padding
                    const int jc = ok ? jj : 0;
                    const size_t idx = (size_t)jc * C_IN + ch;
                    const float va = xa[idx];
                    const float vd = xa[idx + C_HALF];
                    Ba[s] = ok ? va : 0.0f;
                    Bd[s] = ok ? vd : 0.0f;
                }
                acc = __builtin_amdgcn_wmma_f32_16x16x4_f32(
                    false, Alo[c], false, Ba, (short)0, acc, false, false);
                acc = __builtin_amdgcn_wmma_f32_16x16x4_f32(
                    false, Ahi[c], false, Bd, (short)0, acc, false, false);
            }
#pragma unroll
            for (int v = 0; v < 8; ++v) po[co + v * 64] = acc[v];
        }
    }
}

extern "C" void kernel_launch(void* const* d_in, const int* in_sizes, int n_in,
                              void* d_out, int out_size, void* d_ws, size_t ws_size,
                              hipStream_t stream) {
    const float* x      = (const float*)d_in[0];
    const float* rec_lo = (const float*)d_in[1];
    const float* rec_hi = (const float*)d_in[2];
    float* out = (float*)d_out;

    // 16 batches * 1024 w-tiles = 16384 waves; 8 waves per 256-thread block.
    dim3 block(256);
    dim3 grid((16 * 1024) / 8);
    idwt_db4_wmma_kernel<<<grid, block, 0, stream>>>(x, rec_lo, rec_hi, out);
}